// GCN_26594437497094
// MI455X (gfx1250) — compile-verified
//
#include <hip/hip_runtime.h>
#include <hip/hip_bf16.h>
#include <math.h>

typedef __attribute__((ext_vector_type(2))) float v2f;
typedef __attribute__((ext_vector_type(8))) float v8f;

#define N_NODES    50000
#define N_EDGES    800000
#define NUM_GRAPHS 512
#define DIM        128

// ---------------------------------------------------------------------------
// Small elementwise helpers
// ---------------------------------------------------------------------------
__global__ void k_fill(float* __restrict__ p, float v, int n) {
    int i = blockIdx.x * blockDim.x + threadIdx.x;
    if (i < n) p[i] = v;
}

__global__ void k_zero(float* __restrict__ p, unsigned n) {
    unsigned i = blockIdx.x * blockDim.x + threadIdx.x;
    if (i < n) p[i] = 0.0f;
}

__global__ void k_rsqrt_inplace(float* __restrict__ p, int n) {
    int i = blockIdx.x * blockDim.x + threadIdx.x;
    if (i < n) p[i] = rsqrtf(p[i]);
}

// deg[dst] += 1 for every real edge (self loop pre-seeded via k_fill(1.0))
__global__ void k_degree(const int* __restrict__ dst, float* __restrict__ deg) {
    int e = blockIdx.x * blockDim.x + threadIdx.x;
    if (e < N_EDGES) atomicAdd(&deg[dst[e]], 1.0f);
}

// ---------------------------------------------------------------------------
// WMMA f32 GEMM:  C[M x 128] = A[M x 128] * B[128 x 128]
// One wave computes one 16x16 tile of C; 8 waves/block cover N=128.
// V_WMMA_F32_16X16X4_F32 fragment layouts (ISA 7.12.2):
//   A 16x4 : lane<16 -> M=lane, v0=K0,v1=K1 ; lane>=16 -> M=lane-16, v0=K2,v1=K3
//   B 4x16 : lane<16 -> N=lane, v0=K0,v1=K1 ; lane>=16 -> N=lane-16, v0=K2,v1=K3
//   C      : lane<16 -> N=lane, vr = M=r    ; lane>=16 -> N=lane-16, vr = M=r+8
// ---------------------------------------------------------------------------
__global__ __launch_bounds__(256) void k_gemm_wmma(const float* __restrict__ A,
                                                   const float* __restrict__ B,
                                                   float* __restrict__ C) {
    const int wave = threadIdx.x >> 5;     // 0..7  -> N tile
    const int lane = threadIdx.x & 31;
    const int half = lane >> 4;            // 0 or 1 (K sub-block)
    const int idx  = lane & 15;            // M index (A) / N index (B,C)
    const int row0 = blockIdx.x * 16;      // 50000/16 = 3125 exact
    const int n0   = wave * 16;

    const float* __restrict__ arow = A + (size_t)(row0 + idx) * DIM;

    v8f acc = {};
    #pragma unroll 4
    for (int k = 0; k < DIM; k += 4) {
        const int kk = k + half * 2;
        v2f a, b;
        a.x = arow[kk];
        a.y = arow[kk + 1];
        b.x = B[(size_t)kk       * DIM + n0 + idx];
        b.y = B[(size_t)(kk + 1) * DIM + n0 + idx];
        acc = __builtin_amdgcn_wmma_f32_16x16x4_f32(
            /*neg_a=*/false, a, /*neg_b=*/false, b,
            /*c_mod=*/(short)0, acc, /*reuse_a=*/false, /*reuse_b=*/false);
    }

    const int mbase = row0 + half * 8;
    const int col   = n0 + idx;
    #pragma unroll
    for (int r = 0; r < 8; ++r)
        C[(size_t)(mbase + r) * DIM + col] = acc[r];
}

// ---------------------------------------------------------------------------
// Edge message passing: agg[dst] += norm(e) * h[src], one wave32 per edge,
// each lane moves a float4 (32 lanes x 4 = 128 features, coalesced 512B).
// ---------------------------------------------------------------------------
__global__ __launch_bounds__(256) void k_scatter_edges(const float* __restrict__ h,
                                                       const int* __restrict__ src,
                                                       const int* __restrict__ dst,
                                                       const float* __restrict__ dinv,
                                                       float* __restrict__ agg) {
    const int e = blockIdx.x * 8 + (threadIdx.x >> 5);
    if (e >= N_EDGES) return;
    const int lane = threadIdx.x & 31;
    const int s = src[e];
    const int d = dst[e];
    const float norm = dinv[s] * dinv[d];

    const float4 v = ((const float4*)(h + (size_t)s * DIM))[lane];
    float* __restrict__ ap = agg + (size_t)d * DIM + lane * 4;
    atomicAdd(ap + 0, v.x * norm);
    atomicAdd(ap + 1, v.y * norm);
    atomicAdd(ap + 2, v.z * norm);
    atomicAdd(ap + 3, v.w * norm);
}

// out = relu(agg + dinv[n]^2 * h + b)   (self-loop folded in, no atomics)
__global__ void k_bias_relu(const float* __restrict__ agg,
                            const float* __restrict__ h,
                            const float* __restrict__ b,
                            const float* __restrict__ dinv,
                            float* __restrict__ out) {
    const unsigned i = blockIdx.x * blockDim.x + threadIdx.x;
    if (i >= (unsigned)N_NODES * DIM) return;
    const int n = i >> 7;          // /128
    const int d = i & (DIM - 1);
    const float di = dinv[n];
    const float v = agg[i] + di * di * h[i] + b[d];
    out[i] = v > 0.0f ? v : 0.0f;
}

// Segment-sum pooling: wave per node, float4 per lane + one count atomic.
__global__ __launch_bounds__(256) void k_pool(const float* __restrict__ h,
                                              const int* __restrict__ batch,
                                              float* __restrict__ psum,
                                              float* __restrict__ cnt) {
    const int n = blockIdx.x * 8 + (threadIdx.x >> 5);
    if (n >= N_NODES) return;
    const int lane = threadIdx.x & 31;
    const int g = batch[n];
    const float4 v = ((const float4*)(h + (size_t)n * DIM))[lane];
    float* __restrict__ pp = psum + (size_t)g * DIM + lane * 4;
    atomicAdd(pp + 0, v.x);
    atomicAdd(pp + 1, v.y);
    atomicAdd(pp + 2, v.z);
    atomicAdd(pp + 3, v.w);
    if (lane == 0) atomicAdd(&cnt[g], 1.0f);
}

// out[g] = sigmoid( dot(psum[g], Wl) / max(cnt[g],1) + bl )
__global__ __launch_bounds__(256) void k_final(const float* __restrict__ psum,
                                               const float* __restrict__ cnt,
                                               const float* __restrict__ Wl,
                                               const float* __restrict__ bl,
                                               float* __restrict__ out) {
    const int g = blockIdx.x * 8 + (threadIdx.x >> 5);
    if (g >= NUM_GRAPHS) return;
    const int lane = threadIdx.x & 31;

    float acc = 0.0f;
    #pragma unroll
    for (int j = 0; j < 4; ++j) {
        const int d = lane * 4 + j;
        acc += psum[(size_t)g * DIM + d] * Wl[d];
    }
    #pragma unroll
    for (int off = 16; off > 0; off >>= 1)
        acc += __shfl_xor(acc, off, 32);

    if (lane == 0) {
        float c = cnt[g];
        if (c < 1.0f) c = 1.0f;
        const float z = acc / c + bl[0];
        out[g] = 1.0f / (1.0f + __expf(-z));
    }
}

// ---------------------------------------------------------------------------
// Host-side orchestration (all on `stream`, graph-capture safe)
// ---------------------------------------------------------------------------
extern "C" void kernel_launch(void* const* d_in, const int* in_sizes, int n_in,
                              void* d_out, int out_size, void* d_ws, size_t ws_size,
                              hipStream_t stream) {
    const float* x   = (const float*)d_in[0];                 // [N, 128]
    const int*   ei  = (const int*)d_in[1];                   // [2, E]
    const int*   src = ei;
    const int*   dst = ei + N_EDGES;
    const int*   batch = (const int*)d_in[2];                 // [N]
    const float* W1 = (const float*)d_in[3];
    const float* b1 = (const float*)d_in[4];
    const float* W2 = (const float*)d_in[5];
    const float* b2 = (const float*)d_in[6];
    const float* Wl = (const float*)d_in[7];
    const float* bl = (const float*)d_in[8];
    float* out = (float*)d_out;                               // [512]

    // workspace layout (floats); all 16B-aligned
    float* bufA = (float*)d_ws;                               // N*128
    float* bufB = bufA + (size_t)N_NODES * DIM;               // N*128
    float* dinv = bufB + (size_t)N_NODES * DIM;               // N   (deg -> dinv)
    float* psum = dinv + N_NODES;                             // 512*128
    float* cnt  = psum + (size_t)NUM_GRAPHS * DIM;            // 512

    const unsigned NF = (unsigned)N_NODES * DIM;              // 6,400,000
    const int T = 256;
    const int gNodes = (N_NODES + T - 1) / T;                 // 196
    const int gEdges = (N_EDGES + T - 1) / T;                 // 3125
    const int gFeat  = (int)((NF + T - 1) / T);               // 25000
    const int gEdgeW = (N_EDGES + 7) / 8;                     // 100000 (wave/edge)
    const int gNodeW = (N_NODES + 7) / 8;                     // 6250
    const int gGemm  = N_NODES / 16;                          // 3125

    // --- degrees / normalization ---
    k_fill<<<gNodes, T, 0, stream>>>(dinv, 1.0f, N_NODES);                    // self loop
    k_zero<<<(NUM_GRAPHS * DIM + NUM_GRAPHS + T - 1) / T, T, 0, stream>>>(
        psum, (unsigned)(NUM_GRAPHS * DIM + NUM_GRAPHS));                     // psum+cnt (contiguous)
    k_degree<<<gEdges, T, 0, stream>>>(dst, dinv);
    k_rsqrt_inplace<<<gNodes, T, 0, stream>>>(dinv, N_NODES);

    // --- layer 1: h1 = relu(Agg(x@W1) + b1) ---
    k_zero<<<gFeat, T, 0, stream>>>(bufB, NF);
    k_gemm_wmma<<<gGemm, T, 0, stream>>>(x, W1, bufA);                        // bufA = x@W1
    k_scatter_edges<<<gEdgeW, T, 0, stream>>>(bufA, src, dst, dinv, bufB);    // bufB = edge agg
    k_bias_relu<<<gFeat, T, 0, stream>>>(bufB, bufA, b1, dinv, bufA);         // bufA = h1

    // --- layer 2: h2 = relu(Agg(h1@W2) + b2) ---
    k_gemm_wmma<<<gGemm, T, 0, stream>>>(bufA, W2, bufB);                     // bufB = h1@W2
    k_zero<<<gFeat, T, 0, stream>>>(bufA, NF);
    k_scatter_edges<<<gEdgeW, T, 0, stream>>>(bufB, src, dst, dinv, bufA);    // bufA = edge agg
    k_bias_relu<<<gFeat, T, 0, stream>>>(bufA, bufB, b2, dinv, bufB);         // bufB = h2

    // --- pool + head ---
    k_pool<<<gNodeW, T, 0, stream>>>(bufB, batch, psum, cnt);
    k_final<<<(NUM_GRAPHS + 7) / 8, T, 0, stream>>>(psum, cnt, Wl, bl, out);
}